// GHMM_23149873726038
// MI455X (gfx1250) — compile-verified
//
#include <hip/hip_runtime.h>
#include <hip/hip_bf16.h>
#include <math.h>

typedef float v2f __attribute__((ext_vector_type(2)));
typedef float v8f __attribute__((ext_vector_type(8)));

#define BATCH  8
#define NPTS   16384
#define JC     128
#define NSPLIT 32
#define CHUNK  32
#define NPB    (NPTS / NSPLIT)   // 512 n per block

// ---------------------------------------------------------------------------
// Kernel 1: fused stats GEMM.  Each (ns, jpair, b, set) wave computes
//   T[j0+0..31][c=0..7] = sum_{n in chunk} gamma[b,n,j] * X[b,n,c]
// with X = [1, px, py, pz, |p|^2, nx, ny, nz], via V_WMMA_F32_16X16X4_F32.
// Two 16-j tiles per wave (B fragment reused), register-level double buffering
// of the global->LDS staging, zero-padded records (no hot-loop cndmask).
// ---------------------------------------------------------------------------
__global__ __launch_bounds__(32) void ghmm_stats_partial(
    const float* __restrict__ gamma1, const float* __restrict__ pts1,
    const float* __restrict__ pn1,    const float* __restrict__ gamma2,
    const float* __restrict__ pts2,   const float* __restrict__ pn2,
    float* __restrict__ partial)
{
    __shared__ __attribute__((aligned(16))) float ldsG[CHUNK][36]; // 32 j + pad
    __shared__ __attribute__((aligned(16))) float ldsR[CHUNK][20]; // 8 stats + 8 zeros + pad

    const int lane = threadIdx.x;
    const int ns = blockIdx.x;
    const int jt = blockIdx.y;          // 0..3 (pair of 16-j tiles)
    const int bz = blockIdx.z;
    const int b = bz >> 1, s = bz & 1;

    const float* __restrict__ G = s ? gamma2 : gamma1;
    const float* __restrict__ P = s ? pts2   : pts1;
    const float* __restrict__ Q = s ? pn2    : pn1;

    const int j0     = jt * 32;
    const int nbase0 = ns * NPB;
    const int jl     = lane & 15;           // M (j) index for A, N (stat) for B
    const int klo    = (lane >> 4) << 1;    // K sub-offset 0 or 2

    // register staging (double buffer vs. LDS)
    float4 gbuf[8];
    float px, py, pz, qx, qy, qz;

    auto prefetch = [&](int nb) {
#pragma unroll
        for (int t = 0; t < 8; ++t) {
            const int q  = t * 32 + lane;   // 0..255
            const int r  = q >> 3;          // n row 0..31
            const int cc = (q & 7) * 4;     // j sub-col {0,4,...,28}
            gbuf[t] = *(const float4*)(G + ((size_t)b * NPTS + nb + r) * JC + j0 + cc);
        }
        const float* pp = P + ((size_t)b * NPTS + nb + lane) * 3;
        const float* qq = Q + ((size_t)b * NPTS + nb + lane) * 3;
        px = pp[0]; py = pp[1]; pz = pp[2];
        qx = qq[0]; qy = qq[1]; qz = qq[2];
    };

    v8f acc0 = {0.f, 0.f, 0.f, 0.f, 0.f, 0.f, 0.f, 0.f};
    v8f acc1 = {0.f, 0.f, 0.f, 0.f, 0.f, 0.f, 0.f, 0.f};

    prefetch(nbase0);
    for (int c0 = 0; c0 < NPB; c0 += CHUNK) {
        // --- drain staged registers into LDS -------------------------------
#pragma unroll
        for (int t = 0; t < 8; ++t) {
            const int q  = t * 32 + lane;
            const int r  = q >> 3;
            const int cc = (q & 7) * 4;
            *(float4*)(&ldsG[r][cc]) = gbuf[t];
        }
        {
            float4 r0 = {1.0f, px, py, pz};
            float4 r1 = {px*px + py*py + pz*pz, qx, qy, qz};
            float4 zz = {0.0f, 0.0f, 0.0f, 0.0f};
            *(float4*)(&ldsR[lane][0])  = r0;
            *(float4*)(&ldsR[lane][4])  = r1;
            *(float4*)(&ldsR[lane][8])  = zz;   // zero-padded B cols 8..15
            *(float4*)(&ldsR[lane][12]) = zz;
        }
        __syncthreads();

        // --- issue next chunk's global loads (overlap with WMMA) -----------
        if (c0 + CHUNK < NPB) prefetch(nbase0 + c0 + CHUNK);

        // --- WMMA K-march: 16 x (16x16x4 f32) per chunk --------------------
#pragma unroll
        for (int k0 = 0; k0 < CHUNK; k0 += 4) {
            v2f a0, a1, bb;
            a0.x = ldsG[k0 + klo + 0][jl];
            a0.y = ldsG[k0 + klo + 1][jl];
            a1.x = ldsG[k0 + klo + 0][16 + jl];
            a1.y = ldsG[k0 + klo + 1][16 + jl];
            bb.x = ldsR[k0 + klo + 0][jl];      // cols 8..15 already zero
            bb.y = ldsR[k0 + klo + 1][jl];
            acc0 = __builtin_amdgcn_wmma_f32_16x16x4_f32(
                false, a0, false, bb, (short)0, acc0, false, false);
            acc1 = __builtin_amdgcn_wmma_f32_16x16x4_f32(
                false, a1, false, bb, (short)0, acc1, false, false);
        }
        __syncthreads();
    }

    // --- write two 16x8 partial tiles (D layout: lane = col + 16*(M>=8)) ---
    if (jl < 8) {
        const int mbase = (lane >> 4) * 8;
#pragma unroll
        for (int t = 0; t < 2; ++t) {
            float* outp = partial +
                ((((size_t)ns * BATCH + b) * 2 + s) * 8 + (jt * 2 + t)) * 128;
            const v8f a = t ? acc1 : acc0;
#pragma unroll
            for (int v = 0; v < 8; ++v)
                outp[(mbase + v) * 8 + jl] = a[v];
        }
    }
}

// ---------------------------------------------------------------------------
// Kernel 2: reduce over NSPLIT partials -> stats[(b*2+s)*1024 + j*8 + c]
// ---------------------------------------------------------------------------
__global__ __launch_bounds__(256) void ghmm_reduce(
    const float* __restrict__ partial, float* __restrict__ stats)
{
    const int i = blockIdx.x * 256 + threadIdx.x;   // 0..16383
    float acc = 0.0f;
#pragma unroll
    for (int ns = 0; ns < NSPLIT; ++ns)
        acc += partial[(size_t)ns * (BATCH * 2 * JC * 8) + i];
    stats[i] = acc;
}

// ---------------------------------------------------------------------------
// Kernel 3: per-batch finalize: mu/sigma/kappa, H, 3x3 SVD, R, t
// ---------------------------------------------------------------------------
__device__ inline float det3(const float* M) {
    return M[0]*(M[4]*M[8]-M[5]*M[7])
         - M[1]*(M[3]*M[8]-M[5]*M[6])
         + M[2]*(M[3]*M[7]-M[4]*M[6]);
}

__global__ __launch_bounds__(128) void ghmm_finalize(
    const float* __restrict__ stats, float* __restrict__ out)
{
    const int b = blockIdx.x;
    const int j = threadIdx.x;          // 0..127 == J
    __shared__ float red[128][17];

    float T1[8], T2[8];
    const float* s1 = stats + ((size_t)b * 2 + 0) * 1024 + j * 8;
    const float* s2 = stats + ((size_t)b * 2 + 1) * 1024 + j * 8;
#pragma unroll
    for (int c = 0; c < 8; ++c) { T1[c] = s1[c]; T2[c] = s2[c]; }

    const float d1 = T1[0];
    const float d  = d1 + T2[0];
    float mu[3];
#pragma unroll
    for (int a = 0; a < 3; ++a) mu[a] = (T1[1 + a] + T2[1 + a]) / d;
    const float mumu  = mu[0]*mu[0] + mu[1]*mu[1] + mu[2]*mu[2];
    const float sigma = (T1[4] + T2[4] - mumu * d) / (3.0f * d);
    float sv[3];
#pragma unroll
    for (int a = 0; a < 3; ++a) sv[a] = T1[5 + a] + T2[5 + a];
    const float slen  = sqrtf(sv[0]*sv[0] + sv[1]*sv[1] + sv[2]*sv[2]);
    const float kappa = slen / d;
    float mun[3];
#pragma unroll
    for (int a = 0; a < 3; ++a) mun[a] = sv[a] / slen;

    const float isg = 1.0f / sigma;
    const float gso = d1 * isg;
    float W[3];
#pragma unroll
    for (int a = 0; a < 3; ++a) W[a] = T1[1 + a] * isg;
    const float ik = 1.0f / kappa;

    float p[16];
#pragma unroll
    for (int a = 0; a < 3; ++a) { p[a] = W[a]; p[3 + a] = mu[a] * gso; }
    p[6] = gso;
#pragma unroll
    for (int r = 0; r < 3; ++r)
#pragma unroll
        for (int c = 0; c < 3; ++c)
            p[7 + r*3 + c] = mu[r] * W[c] + T1[5 + r] * mun[c] * ik;

#pragma unroll
    for (int i = 0; i < 16; ++i) red[j][i] = p[i];
    __syncthreads();
    if (j != 0) return;

    float S[16];
#pragma unroll
    for (int i = 0; i < 16; ++i) S[i] = 0.0f;
    for (int t = 0; t < 128; ++t)
#pragma unroll
        for (int i = 0; i < 16; ++i) S[i] += red[t][i];

    const float mp[3] = {S[0], S[1], S[2]};
    const float mm[3] = {S[3], S[4], S[5]};
    const float sow   = S[6];
    float H[9];
    for (int r = 0; r < 3; ++r)
        for (int c = 0; c < 3; ++c)
            H[r*3 + c] = S[7 + r*3 + c] - mm[r] * mp[c] / sow;

    // ---- SVD(H) via Jacobi on K = H^T H ----
    float K[9];
    for (int i = 0; i < 3; ++i)
        for (int l = 0; l < 3; ++l) {
            float acc = 0.0f;
            for (int k = 0; k < 3; ++k) acc += H[k*3 + i] * H[k*3 + l];
            K[i*3 + l] = acc;
        }
    float V[9] = {1,0,0, 0,1,0, 0,0,1};
    for (int sweep = 0; sweep < 12; ++sweep) {
        const int PP[3] = {0, 0, 1}, QQ[3] = {1, 2, 2};
        for (int pi = 0; pi < 3; ++pi) {
            const int pq = PP[pi], qq2 = QQ[pi];
            const float apq = K[pq*3 + qq2];
            if (fabsf(apq) < 1e-30f) continue;
            const float app = K[pq*3 + pq], aqq = K[qq2*3 + qq2];
            const float tau = (aqq - app) / (2.0f * apq);
            const float tt  = copysignf(1.0f, tau) / (fabsf(tau) + sqrtf(1.0f + tau*tau));
            const float cc  = 1.0f / sqrtf(1.0f + tt*tt);
            const float ss  = tt * cc;
            for (int k = 0; k < 3; ++k) {              // K <- K J
                const float kp = K[k*3 + pq], kq = K[k*3 + qq2];
                K[k*3 + pq] = cc*kp - ss*kq; K[k*3 + qq2] = ss*kp + cc*kq;
            }
            for (int k = 0; k < 3; ++k) {              // K <- J^T K
                const float kp = K[pq*3 + k], kq = K[qq2*3 + k];
                K[pq*3 + k] = cc*kp - ss*kq; K[qq2*3 + k] = ss*kp + cc*kq;
            }
            for (int k = 0; k < 3; ++k) {              // V <- V J
                const float vp = V[k*3 + pq], vq = V[k*3 + qq2];
                V[k*3 + pq] = cc*vp - ss*vq; V[k*3 + qq2] = ss*vp + cc*vq;
            }
        }
    }
    float w[3] = {K[0], K[4], K[8]};
    for (int i = 0; i < 2; ++i)                        // sort descending
        for (int l = i + 1; l < 3; ++l)
            if (w[l] > w[i]) {
                const float tw = w[i]; w[i] = w[l]; w[l] = tw;
                for (int k = 0; k < 3; ++k) {
                    const float tv = V[k*3 + i]; V[k*3 + i] = V[k*3 + l]; V[k*3 + l] = tv;
                }
            }
    float U[9];
    for (int i = 0; i < 3; ++i) {
        float u[3];
        for (int r = 0; r < 3; ++r)
            u[r] = H[r*3+0]*V[0*3+i] + H[r*3+1]*V[1*3+i] + H[r*3+2]*V[2*3+i];
        const float nrm = sqrtf(u[0]*u[0] + u[1]*u[1] + u[2]*u[2]);
        if (nrm > 1e-20f) {
            for (int r = 0; r < 3; ++r) U[r*3 + i] = u[r] / nrm;
        } else if (i == 2) {                            // degenerate: cross col0 x col1
            U[0*3+2] = U[1*3+0]*U[2*3+1] - U[2*3+0]*U[1*3+1];
            U[1*3+2] = U[2*3+0]*U[0*3+1] - U[0*3+0]*U[2*3+1];
            U[2*3+2] = U[0*3+0]*U[1*3+1] - U[1*3+0]*U[0*3+1];
        } else {
            for (int r = 0; r < 3; ++r) U[r*3 + i] = (r == i) ? 1.0f : 0.0f;
        }
    }
    const float dd = (det3(U) * det3(V) < 0.0f) ? -1.0f : 1.0f;
    const float dg[3] = {1.0f, 1.0f, dd};
    float R[9];
    for (int r = 0; r < 3; ++r)
        for (int c = 0; c < 3; ++c) {
            float acc = 0.0f;
            for (int k = 0; k < 3; ++k) acc += V[r*3 + k] * dg[k] * U[c*3 + k];
            R[r*3 + c] = acc;
        }
    float t3[3];
    for (int r = 0; r < 3; ++r)
        t3[r] = (mm[r] - (R[r*3+0]*mp[0] + R[r*3+1]*mp[1] + R[r*3+2]*mp[2])) / sow;

    for (int i = 0; i < 9; ++i) out[b*9 + i] = R[i];
    for (int i = 0; i < 3; ++i) out[BATCH*9 + b*3 + i] = t3[i];
}

// ---------------------------------------------------------------------------
extern "C" void kernel_launch(void* const* d_in, const int* in_sizes, int n_in,
                              void* d_out, int out_size, void* d_ws, size_t ws_size,
                              hipStream_t stream) {
    (void)in_sizes; (void)n_in; (void)out_size; (void)ws_size;
    const float* gamma1 = (const float*)d_in[0];
    const float* pts1   = (const float*)d_in[1];
    const float* pn1    = (const float*)d_in[2];
    const float* gamma2 = (const float*)d_in[3];
    const float* pts2   = (const float*)d_in[4];
    const float* pn2    = (const float*)d_in[5];
    float* out = (float*)d_out;

    float* partial = (float*)d_ws;                                  // 524288 f32
    float* stats   = partial + (size_t)NSPLIT * BATCH * 2 * JC * 8; // 16384 f32

    dim3 g1(NSPLIT, JC / 32, BATCH * 2);
    ghmm_stats_partial<<<g1, 32, 0, stream>>>(gamma1, pts1, pn1,
                                              gamma2, pts2, pn2, partial);
    ghmm_reduce<<<(BATCH * 2 * JC * 8) / 256, 256, 0, stream>>>(partial, stats);
    ghmm_finalize<<<BATCH, 128, 0, stream>>>(stats, out);
}